// SpatialGraphBuilder4_21277267984967
// MI455X (gfx1250) — compile-verified
//
#include <hip/hip_runtime.h>
#include <hip/hip_bf16.h>
#include <stdint.h>

typedef __attribute__((ext_vector_type(2))) float v2f;
typedef __attribute__((ext_vector_type(8))) float v8f;

#define GLOBAL_AS __attribute__((address_space(1)))
#define LDS_AS    __attribute__((address_space(3)))

#define NDIM 2048
#define NN   (NDIM*NDIM)
#define CIN  8
#define KTOT 72
#define KSTEPS 18
#define EDGE_K 32768u
#define EQCAP  65536u

#define TX 64
#define TY 16
#define HX 66
#define HY 18

// workspace layout (32-bit word offsets)
#define OFF_RAW  0ull
#define OFF_KEYS (8ull*NN)
#define OFF_H1   (OFF_KEYS + (unsigned long long)NN)
#define OFF_H2   (OFF_H1 + 65536ull)
#define OFF_META (OFF_H2 + 65536ull)
#define OFF_CGK  (OFF_META + 64ull)
#define OFF_CGI  (OFF_CGK + EDGE_K)
#define OFF_CE   (OFF_CGI + EDGE_K)
#define OFF_FIN  (OFF_CE + EQCAP)   // u64 region, 2*EDGE_K words

// LDS flat offset for im2col position kk = c*9 + ky*3 + kx (constexpr-folded)
__host__ __device__ constexpr int offtab(int kk)
{
  return ((kk / 9) * HY + (kk % 9) / 3) * HX + (kk % 9) % 3;
}

// ---------------------------------------------------------------------------
// K1: fused dual conv3x3 (8 out channels) as GEMM via f32 WMMA 16x16x4.
// Interior tiles stage input via async global->LDS (ASYNCcnt path); border
// tiles use the predicated sync path with zero fill.
// ---------------------------------------------------------------------------
__global__ __launch_bounds__(256)
void conv_score_wmma(const float* __restrict__ rel,
                     const float* __restrict__ w0, const float* __restrict__ b0,
                     const float* __restrict__ w1, const float* __restrict__ b1,
                     float* __restrict__ rawp)
{
  __shared__ float lin[CIN][HY][HX];     // input tile + halo
  __shared__ float wmat[KTOT][16];       // B matrix [72 x 16] (cols 8..15 zero)
  const int tid = threadIdx.x;
  const int bx = blockIdx.x, by = blockIdx.y;

  // stage weight matrix: row kk = c*9+ky*3+kx, col e = output channel
  for (int t = tid; t < KTOT * 16; t += 256) {
    int kk = t >> 4, e = t & 15;
    float v = 0.f;
    if (e < 8) v = ((e < 4) ? w0 : w1)[(e & 3) * KTOT + kk];
    wmat[kk][e] = v;
  }

  // stage input tile with 1-pixel halo
  const int gx0 = bx * TX - 1, gy0 = by * TY - 1;
  const bool interior =
      (bx >= 1) && (bx <= (NDIM / TX) - 2) && (by >= 1) && (by <= (NDIM / TY) - 2);
#if __has_builtin(__builtin_amdgcn_global_load_async_to_lds_b32)
  if (interior) {
    // direct global->LDS async copy: no VGPR round-trip, no ds_store traffic
    for (int t = tid; t < CIN * HY * HX; t += 256) {
      int c = t / (HY * HX), r = (t / HX) % HY, x = t % HX;
      const float* src = rel + (size_t)c * NN + (size_t)(gy0 + r) * NDIM + (gx0 + x);
      __builtin_amdgcn_global_load_async_to_lds_b32(
          (GLOBAL_AS int*)src, (LDS_AS int*)&lin[c][r][x], 0, 0);
    }
    asm volatile("s_wait_asynccnt 0" ::: "memory");
  } else
#endif
  {
    for (int t = tid; t < CIN * HY * HX; t += 256) {
      int c = t / (HY * HX), r = (t / HX) % HY, x = t % HX;
      int gy = gy0 + r, gx = gx0 + x;
      float v = 0.f;
      if ((unsigned)gy < (unsigned)NDIM && (unsigned)gx < (unsigned)NDIM)
        v = rel[(size_t)c * NN + (size_t)gy * NDIM + gx];
      lin[c][r][x] = v;
    }
  }
  __syncthreads();

  const int lane  = tid & 31, wave = tid >> 5;
  const int m     = lane & 15;          // pixel within 16-group (A row)
  const int half  = lane >> 4;          // 0: K0/K1 half, 1: K2/K3 half
  const int khalf = half << 1;
  const int col   = lane & 15;          // B column = output channel
  const bool hi   = (half != 0);

  // per-lane B fragments + LDS A-offsets: every offset is one of two
  // compile-time constants, selected once per lane (v_cndmask), so the hot
  // loop is pure base+off ds_loads feeding WMMA.
  v2f bfrag[KSTEPS];
  int aoff0[KSTEPS], aoff1[KSTEPS];
  #pragma unroll
  for (int s = 0; s < KSTEPS; ++s) {
    bfrag[s].x = wmat[s * 4 + khalf][col];
    bfrag[s].y = wmat[s * 4 + khalf + 1][col];
    aoff0[s] = hi ? offtab(s * 4 + 2) : offtab(s * 4 + 0);
    aoff1[s] = hi ? offtab(s * 4 + 3) : offtab(s * 4 + 1);
  }
  const float* linf = &lin[0][0][0];
  const int e = col;
  const float be = (e < 4) ? b0[e] : ((e < 8) ? b1[e - 4] : 0.f);

  for (int rr = 0; rr < 2; ++rr) {
    const int yloc = wave * 2 + rr;     // output row within tile
    for (int g = 0; g < 4; ++g) {
      const int base = yloc * HX + g * 16 + m;
      v8f acc = {};
      #pragma unroll
      for (int s = 0; s < KSTEPS; ++s) {
        v2f a;
        a.x = linf[base + aoff0[s]];
        a.y = linf[base + aoff1[s]];
        acc = __builtin_amdgcn_wmma_f32_16x16x4_f32(
            false, a, false, bfrag[s], (short)0, acc, false, false);
      }
      if (e < 8) {
        // D layout: lane<16 holds pixels 0..7 (vgpr v), lane>=16 pixels 8..15
        const int gy  = by * TY + yloc;
        const int gxb = bx * TX + g * 16 + half * 8;
        float* op = rawp + (size_t)e * NN + (size_t)gy * NDIM + gxb;
        float4 lo = make_float4(acc[0] + be, acc[1] + be, acc[2] + be, acc[3] + be);
        float4 hi4 = make_float4(acc[4] + be, acc[5] + be, acc[6] + be, acc[7] + be);
        *(float4*)(op)     = lo;
        *(float4*)(op + 4) = hi4;
      }
    }
  }
}

// ---------------------------------------------------------------------------
// K0: zero radix histograms + meta/counters (re-run every call: deterministic)
// ---------------------------------------------------------------------------
__global__ void zero_meta(unsigned* __restrict__ h1)
{
  unsigned i = blockIdx.x * 256u + threadIdx.x;
  if (i < 65536u + 65536u + 64u) h1[i] = 0u;
}

// ---------------------------------------------------------------------------
// K2: weights(i,j) from symmetric score (raw fits in 192MB L2, transposed
// reads hit L2), sortable key, high-16 radix histogram.
// ---------------------------------------------------------------------------
__global__ __launch_bounds__(256)
void hist_pass1(const float* __restrict__ rawp,
                const float* __restrict__ wsw, const float* __restrict__ bsw,
                unsigned* __restrict__ keys, unsigned* __restrict__ h1)
{
  unsigned p = blockIdx.x * 256u + threadIdx.x;
  unsigned i = p >> 11, j = p & 2047u;
  float s = bsw[1] - bsw[0];
  #pragma unroll
  for (int c = 0; c < 8; ++c) {
    float dw = wsw[8 + c] - wsw[c];
    s += dw * 0.5f * (rawp[(size_t)c * NN + ((size_t)i << 11) + j]
                    + rawp[(size_t)c * NN + ((size_t)j << 11) + i]);
  }
  // adj = (s > 0); masked pixels get key 0 (below every valid key)
  unsigned key = (s > 0.f) ? (__float_as_uint(s) | 0x80000000u) : 0u;
  keys[p] = key;
  atomicAdd(&h1[key >> 16], 1u);
}

__global__ void scan_pass1(const unsigned* __restrict__ h1, unsigned* __restrict__ meta)
{
  if (threadIdx.x != 0 || blockIdx.x != 0) return;
  unsigned cum = 0; int b = 65535;
  for (; b > 0; --b) { unsigned h = h1[b]; if (cum + h >= EDGE_K) break; cum += h; }
  meta[0] = (unsigned)b;   // threshold high-16 bin
  meta[1] = cum;           // count strictly above bin
}

__global__ __launch_bounds__(256)
void hist_pass2(const unsigned* __restrict__ keys, const unsigned* __restrict__ meta,
                unsigned* __restrict__ h2)
{
  unsigned p = blockIdx.x * 256u + threadIdx.x;
  unsigned B = meta[0];
  unsigned key = keys[p];
  if ((key >> 16) == B) atomicAdd(&h2[key & 0xFFFFu], 1u);
}

__global__ void scan_pass2(const unsigned* __restrict__ h2, unsigned* __restrict__ meta)
{
  if (threadIdx.x != 0 || blockIdx.x != 0) return;
  unsigned cum = meta[1]; int b = 65535;
  for (; b > 0; --b) { unsigned h = h2[b]; if (cum + h >= EDGE_K) break; cum += h; }
  meta[2] = (meta[0] << 16) | (unsigned)b;  // exact 32-bit key threshold
  meta[3] = cum;                            // count strictly greater
  meta[4] = EDGE_K - cum;                   // ties needed at keyT
}

__global__ __launch_bounds__(256)
void collect_cands(const unsigned* __restrict__ keys, unsigned* __restrict__ meta,
                   unsigned* __restrict__ cgk, unsigned* __restrict__ cgi,
                   unsigned* __restrict__ ce)
{
  unsigned p = blockIdx.x * 256u + threadIdx.x;
  unsigned key = keys[p];
  unsigned keyT = meta[2];
  if (key > keyT) {
    unsigned s = atomicAdd(&meta[5], 1u);
    if (s < EDGE_K) { cgk[s] = key; cgi[s] = p; }
  } else if (key == keyT) {
    unsigned s = atomicAdd(&meta[6], 1u);
    if (s < EQCAP) ce[s] = p;   // collect ALL ties; order fixed by sort below
  }
}

// ---------------------------------------------------------------------------
// K7: single-workgroup finalize. Sort ties asc by index, build exactly K
// records packed as ((~key)<<32 | idx) so ascending u64 sort == (value desc,
// index asc) -- matches jax top_k ordering. Data lives in ws (global), only
// this block touches it, so __syncthreads suffices between bitonic passes.
// ---------------------------------------------------------------------------
__global__ __launch_bounds__(1024)
void finalize_sort(const unsigned* __restrict__ meta,
                   const unsigned* __restrict__ cgk, const unsigned* __restrict__ cgi,
                   unsigned* __restrict__ ce, unsigned long long* __restrict__ fin)
{
  const unsigned tid = threadIdx.x, NT = 1024;
  const unsigned keyT = meta[2];
  unsigned cntG = meta[5]; if (cntG > EDGE_K) cntG = EDGE_K;
  unsigned cntE = meta[6]; if (cntE > EQCAP) cntE = EQCAP;

  for (unsigned p = tid; p < EQCAP; p += NT)
    if (p >= cntE) ce[p] = 0xFFFFFFFFu;
  __syncthreads();

  // ascending bitonic sort of tie indices
  for (unsigned k = 2; k <= EQCAP; k <<= 1) {
    for (unsigned j = k >> 1; j > 0; j >>= 1) {
      for (unsigned i = tid; i < EQCAP; i += NT) {
        unsigned l = i ^ j;
        if (l > i) {
          unsigned a = ce[i], b = ce[l];
          bool up = ((i & k) == 0);
          if ((a > b) == up) { ce[i] = b; ce[l] = a; }
        }
      }
      __syncthreads();
    }
  }

  for (unsigned t = tid; t < EDGE_K; t += NT) {
    unsigned key, idx;
    if (t < cntG) { key = cgk[t]; idx = cgi[t]; }
    else {
      key = keyT;
      unsigned eix = t - cntG;
      idx = (eix < EQCAP) ? ce[eix] : 0u;
    }
    fin[t] = ((unsigned long long)(~key) << 32) | (unsigned long long)idx;
  }
  __syncthreads();

  for (unsigned k = 2; k <= EDGE_K; k <<= 1) {
    for (unsigned j = k >> 1; j > 0; j >>= 1) {
      for (unsigned i = tid; i < EDGE_K; i += NT) {
        unsigned l = i ^ j;
        if (l > i) {
          unsigned long long a = fin[i], b = fin[l];
          bool up = ((i & k) == 0);
          if ((a > b) == up) { fin[i] = b; fin[l] = a; }
        }
      }
      __syncthreads();
    }
  }
}

// ---------------------------------------------------------------------------
// K8: outputs: edge_index (int bits in float buffer), vals, edge_attr gather.
// ---------------------------------------------------------------------------
__global__ __launch_bounds__(256)
void write_outputs(const unsigned long long* __restrict__ fin,
                   const float* __restrict__ rawp,
                   float* __restrict__ outf)
{
  unsigned t = blockIdx.x * 256u + threadIdx.x;
  if (t >= EDGE_K) return;
  unsigned long long f = fin[t];
  unsigned idx = (unsigned)(f & 0xFFFFFFFFull) & (NN - 1u);
  unsigned key = ~((unsigned)(f >> 32));
  unsigned i = idx >> 11, j = idx & 2047u;
  int* outi = (int*)outf;
  outi[t] = (int)i;
  outi[EDGE_K + t] = (int)j;
  outf[2u * EDGE_K + t] = key ? __uint_as_float(key & 0x7FFFFFFFu) : -1e30f;
  #pragma unroll
  for (int c = 0; c < 8; ++c)
    outf[3u * EDGE_K + (size_t)t * 8 + c] =
        0.5f * (rawp[(size_t)c * NN + ((size_t)i << 11) + j]
              + rawp[(size_t)c * NN + ((size_t)j << 11) + i]);
}

// ---------------------------------------------------------------------------
extern "C" void kernel_launch(void* const* d_in, const int* in_sizes, int n_in,
                              void* d_out, int out_size, void* d_ws, size_t ws_size,
                              hipStream_t stream)
{
  (void)in_sizes; (void)n_in; (void)out_size; (void)ws_size;
  const float* rel = (const float*)d_in[0];
  const float* w0  = (const float*)d_in[1];
  const float* b0  = (const float*)d_in[2];
  const float* w1  = (const float*)d_in[3];
  const float* b1  = (const float*)d_in[4];
  const float* wsw = (const float*)d_in[5];
  const float* bsw = (const float*)d_in[6];
  float* outf = (float*)d_out;

  unsigned* wsu = (unsigned*)d_ws;
  float* rawp   = (float*)d_ws + OFF_RAW;
  unsigned* keys = wsu + OFF_KEYS;
  unsigned* h1   = wsu + OFF_H1;
  unsigned* h2   = wsu + OFF_H2;
  unsigned* meta = wsu + OFF_META;
  unsigned* cgk  = wsu + OFF_CGK;
  unsigned* cgi  = wsu + OFF_CGI;
  unsigned* ce   = wsu + OFF_CE;
  unsigned long long* fin = (unsigned long long*)(wsu + OFF_FIN);

  zero_meta<<<(65536u + 65536u + 64u + 255u) / 256u, 256, 0, stream>>>(h1);
  conv_score_wmma<<<dim3(NDIM / TX, NDIM / TY), 256, 0, stream>>>(rel, w0, b0, w1, b1, rawp);
  hist_pass1<<<NN / 256, 256, 0, stream>>>(rawp, wsw, bsw, keys, h1);
  scan_pass1<<<1, 1, 0, stream>>>(h1, meta);
  hist_pass2<<<NN / 256, 256, 0, stream>>>(keys, meta, h2);
  scan_pass2<<<1, 1, 0, stream>>>(h2, meta);
  collect_cands<<<NN / 256, 256, 0, stream>>>(keys, meta, cgk, cgi, ce);
  finalize_sort<<<1, 1024, 0, stream>>>(meta, cgk, cgi, ce, fin);
  write_outputs<<<(EDGE_K + 255u) / 256u, 256, 0, stream>>>(fin, rawp, outf);
}